// xLSTM_46978352284245
// MI455X (gfx1250) — compile-verified
//
#include <hip/hip_runtime.h>
#include <math.h>

// ---------------------------------------------------------------------------
// xLSTM forward for MI455X (gfx1250, wave32). fp32 end-to-end; all GEMMs use
// V_WMMA_F32_16X16X4_F32 (fp32 A/B, fp32 accumulate) so numerics match the
// fp32 JAX reference. The whole working set is L2-resident (<=~32MB scratch
// per batch chunk vs 192MB L2) -> bandwidth problem; fp32 WMMA keeps full
// precision at negligible compute cost.
//
// GEMM B-operands: every weight matrix is tiny (<=25.6KB), so each workgroup
// stages the full, zero-padded W into LDS in a (k-pair, col)-interleaved
// layout. Inner loop is then: global_load_b64 (A pair) + ds_load_b64 (B pair)
// + v_wmma -- no divergent guards / exec-mask churn in the hot loop.
// ---------------------------------------------------------------------------

typedef float v2f __attribute__((ext_vector_type(2)));
typedef float v8f __attribute__((ext_vector_type(8)));

#define SEQ 24
#define NHM 8
#define DHM 10
#define NHS 8
#define DHS 5

#define GEMM_LDS_MAX 6400   // max K*Npad over all GEMMs (up_w: 40*160)

// ------------------------------- WMMA GEMM ---------------------------------
// One wave -> one 16x16 tile of D = A[M,K] @ W[K,N], K % 4 == 0, M % 16 == 0.
// fp32 16x16x4 layout: lanes 0-15 hold K={k,k+1}, lanes 16-31 hold K={k+2,k+3}.
// D = A@W (+bias) (+res). bias/res may be null. D may alias res.
__global__ __launch_bounds__(256)
void k_gemm(const float* __restrict__ A, const float* __restrict__ W,
            const float* __restrict__ bias, const float* __restrict__ res,
            float* __restrict__ D, int M, int K, int N, int Ntiles) {
  __shared__ float Ws[GEMM_LDS_MAX];
  int Npad = Ntiles << 4;
  int total = K * Npad;
  // stage W, zero-padded, pair-interleaved: Ws[((kk/2)*Npad+cc)*2 + (kk&1)]
  for (int idx = threadIdx.x; idx < total; idx += blockDim.x) {
    int kk = idx / Npad, cc = idx - kk * Npad;
    float v = (cc < N) ? W[(size_t)kk * N + cc] : 0.0f;
    Ws[(((kk >> 1) * Npad + cc) << 1) | (kk & 1)] = v;
  }
  __syncthreads();

  int wave = (blockIdx.x * blockDim.x + threadIdx.x) >> 5;
  int lane = threadIdx.x & 31;
  int mt = wave / Ntiles, nt = wave - mt * Ntiles;
  if (mt * 16 >= M) return;                        // wave-uniform
  int half = lane >> 4, l15 = lane & 15;
  const float* arow = A + (size_t)(mt * 16 + l15) * K;
  int col = (nt << 4) + l15;                       // < Npad always
  v8f acc = {};
  for (int k = 0; k < K; k += 4) {
    int ka = k + (half << 1);                      // even
    v2f a; a.x = arow[ka]; a.y = arow[ka + 1];     // contiguous 8B
    v2f b = *(const v2f*)&Ws[(((ka >> 1) * Npad) + col) << 1];  // ds b64
    acc = __builtin_amdgcn_wmma_f32_16x16x4_f32(false, a, false, b,
                                                (short)0, acc, false, false);
  }
  if (col < N) {
    float badd = bias ? bias[col] : 0.0f;
    for (int r = 0; r < 8; ++r) {
      int row = mt * 16 + r + half * 8;
      float v = acc[r] + badd;
      if (res) v += res[(size_t)row * N + col];
      D[(size_t)row * N + col] = v;
    }
  }
}

// Gates: G[t,0:8] = qkv@ig_w + ig_b ; G[t,8:16] = qkv@fg_w + fg_b. K=240, N=16.
__global__ __launch_bounds__(256)
void k_gemm_gates(const float* __restrict__ QKV,
                  const float* __restrict__ igw, const float* __restrict__ fgw,
                  const float* __restrict__ igb, const float* __restrict__ fgb,
                  float* __restrict__ G, int M) {
  __shared__ float Ws[240 * 16];
  for (int idx = threadIdx.x; idx < 240 * 16; idx += blockDim.x) {
    int kk = idx >> 4, cc = idx & 15;
    const float* Wp = (cc < 8) ? igw : fgw;
    Ws[((((kk >> 1) << 4) | cc) << 1) | (kk & 1)] = Wp[kk * 8 + (cc & 7)];
  }
  __syncthreads();

  int wave = (blockIdx.x * blockDim.x + threadIdx.x) >> 5;
  int lane = threadIdx.x & 31;
  if (wave * 16 >= M) return;
  int half = lane >> 4, l15 = lane & 15;
  const float* arow = QKV + (size_t)(wave * 16 + l15) * 240;
  int col = l15;
  v8f acc = {};
  for (int k = 0; k < 240; k += 4) {
    int ka = k + (half << 1);
    v2f a; a.x = arow[ka]; a.y = arow[ka + 1];
    v2f b = *(const v2f*)&Ws[((((ka >> 1) << 4) | col) << 1)];
    acc = __builtin_amdgcn_wmma_f32_16x16x4_f32(false, a, false, b,
                                                (short)0, acc, false, false);
  }
  float badd = (col < 8) ? igb[col] : fgb[col & 7];
  for (int r = 0; r < 8; ++r) {
    int row = wave * 16 + r + half * 8;
    G[(size_t)row * 16 + col] = acc[r] + badd;
  }
}

// --------------------------- elementwise kernels ---------------------------
__device__ __forceinline__ float siluf(float v) { return v / (1.0f + expf(-v)); }

// h[t,e] = x[t,:]@in_w[:,e] + in_b[e] + posenc(s,e)
__global__ void k_input(const float* __restrict__ x, const float* __restrict__ in_w,
                        const float* __restrict__ in_b, float* __restrict__ h, int T) {
  int idx = blockIdx.x * blockDim.x + threadIdx.x;
  if (idx >= T * 40) return;
  int t = idx / 40, e = idx % 40;
  const float* xr = x + (size_t)t * 10;
  float acc = in_b[e];
  for (int f = 0; f < 10; ++f) acc += xr[f] * in_w[f * 40 + e];
  int s = t % SEQ;
  float freq = expf(-(float)(e & ~1) * (9.210340371976184f / 40.0f)); // ln(1e4)/E
  float ang = (float)s * freq;
  acc += (e & 1) ? cosf(ang) : sinf(ang);
  h[idx] = acc;
}

// y = LayerNorm(x)*w over last dim E
__global__ void k_layernorm(const float* __restrict__ x, const float* __restrict__ w,
                            float* __restrict__ y, int T, int E) {
  int t = blockIdx.x * blockDim.x + threadIdx.x;
  if (t >= T) return;
  const float* r = x + (size_t)t * E;
  float mu = 0.f;
  for (int e = 0; e < E; ++e) mu += r[e];
  mu /= (float)E;
  float var = 0.f;
  for (int e = 0; e < E; ++e) { float d = r[e] - mu; var += d * d; }
  var /= (float)E;
  float inv = rsqrtf(var + 1e-5f);
  float* o = y + (size_t)t * E;
  for (int e = 0; e < E; ++e) o[e] = (r[e] - mu) * inv * w[e];
}

// depthwise causal conv (K=4) + SiLU.  src has leading-dim ld (>=C).
__global__ void k_conv_silu(const float* __restrict__ src, int ld,
                            const float* __restrict__ cw, const float* __restrict__ cb,
                            float* __restrict__ dst, int T, int C) {
  int idx = blockIdx.x * blockDim.x + threadIdx.x;
  if (idx >= T * C) return;
  int t = idx / C, c = idx % C;
  int s = t % SEQ;
  float acc = cb[c];
#pragma unroll
  for (int j = 0; j < 4; ++j) {
    int sp = s - 3 + j;
    if (sp >= 0) acc += cw[j * C + c] * src[(size_t)(t - 3 + j) * ld + c];
  }
  dst[(size_t)t * C + c] = siluf(acc);
}

// block-diagonal 4x4 q,k (from xc) and v (from xm = up[:,0:80]); qkv[T,240]
__global__ void k_headwise_qkv(const float* __restrict__ xc, const float* __restrict__ up,
                               const float* __restrict__ qw, const float* __restrict__ kw,
                               const float* __restrict__ vw, float* __restrict__ qkv, int T) {
  int idx = blockIdx.x * blockDim.x + threadIdx.x;
  if (idx >= T * 80) return;
  int t = idx / 80, c = idx % 80;
  int blk = c >> 2, o = c & 3;
  const float* xr = xc + (size_t)t * 80 + blk * 4;
  const float* mr = up + (size_t)t * 160 + blk * 4;
  float q = 0.f, k = 0.f, v = 0.f;
#pragma unroll
  for (int i = 0; i < 4; ++i) {
    int wix = blk * 16 + i * 4 + o;
    q += xr[i] * qw[wix];
    k += xr[i] * kw[wix];
    v += mr[i] * vw[wix];
  }
  float* orow = qkv + (size_t)t * 240;
  orow[c] = q; orow[80 + c] = k; orow[160 + c] = v;
}

// one wave per (batch-in-chunk, head): stabilized parallel mLSTM attention
__global__ void k_attn(const float* __restrict__ qkv, const float* __restrict__ gates,
                       float* __restrict__ hatt) {
  __shared__ float qs[SEQ * DHM], ks[SEQ * DHM], vs[SEQ * DHM];
  __shared__ float lsfs[SEQ], igs[SEQ], cum[SEQ];
  int b = blockIdx.x / NHM, hd = blockIdx.x % NHM;
  int lane = threadIdx.x;
  if (lane < SEQ) {
    int t = b * SEQ + lane;
    const float* row = qkv + (size_t)t * 240 + hd * DHM;
    for (int d = 0; d < DHM; ++d) {
      qs[lane * DHM + d] = row[d];
      ks[lane * DHM + d] = row[80 + d];
      vs[lane * DHM + d] = row[160 + d];
    }
    float ig = gates[(size_t)t * 16 + hd];
    float fg = gates[(size_t)t * 16 + 8 + hd];
    igs[lane]  = ig;
    lsfs[lane] = (fg >= 0.f) ? -log1pf(expf(-fg)) : (fg - log1pf(expf(fg)));
  }
  __syncthreads();
  if (lane < SEQ) {
    float c = 0.f;
    for (int t2 = 0; t2 <= lane; ++t2) c += lsfs[t2];
    cum[lane] = c;
  }
  __syncthreads();
  if (lane < SEQ) {
    int s = lane;
    float m = -3.4e38f;
    for (int t2 = 0; t2 <= s; ++t2) {
      float v = cum[s] - cum[t2] + igs[t2];
      m = fmaxf(m, v);
    }
    float rowsum = 0.f;
    float acc[DHM];
    for (int d = 0; d < DHM; ++d) acc[d] = 0.f;
    const float scale = 0.31622776601683794f; // 10^-0.5
    for (int t2 = 0; t2 <= s; ++t2) {
      float dm = expf(cum[s] - cum[t2] + igs[t2] - m);
      float qk = 0.f;
      for (int d = 0; d < DHM; ++d) qk += qs[s * DHM + d] * ks[t2 * DHM + d];
      float Cst = qk * scale * dm;
      rowsum += Cst;
      for (int d = 0; d < DHM; ++d) acc[d] += Cst * vs[t2 * DHM + d];
    }
    float nden = fmaxf(fabsf(rowsum), expf(-m)) + 1e-6f;
    float* o = hatt + (size_t)(b * SEQ + s) * 80 + hd * DHM;
    for (int d = 0; d < DHM; ++d) o[d] = acc[d] / nden;
  }
}

// per-head norm + skip + SiLU(z) gating; in-place safe on hatt
__global__ void k_mhnorm_gate(const float* __restrict__ hatt, const float* __restrict__ mw,
                              const float* __restrict__ skip, const float* __restrict__ xc,
                              const float* __restrict__ up, float* __restrict__ h80, int T) {
  int idx = blockIdx.x * blockDim.x + threadIdx.x;
  if (idx >= T * NHM) return;
  int t = idx / NHM, hd = idx % NHM;
  const float* r = hatt + (size_t)t * 80 + hd * DHM;
  float loc[DHM];
  float mu = 0.f;
  for (int d = 0; d < DHM; ++d) { loc[d] = r[d]; mu += loc[d]; }
  mu *= (1.0f / DHM);
  float var = 0.f;
  for (int d = 0; d < DHM; ++d) { float dd = loc[d] - mu; var += dd * dd; }
  var *= (1.0f / DHM);
  float inv = rsqrtf(var + 1e-5f);
  for (int d = 0; d < DHM; ++d) {
    int c = hd * DHM + d;
    float v = (loc[d] - mu) * inv * mw[c];
    v += skip[c] * xc[(size_t)t * 80 + c];
    float z = up[(size_t)t * 160 + 80 + c];
    h80[(size_t)t * 80 + c] = v * siluf(z);
  }
}

// sLSTM pre-activations: i,f from xc; z,o from xn. pre[T,160] = [i|f|z|o]
__global__ void k_slstm_pre(const float* __restrict__ xc, const float* __restrict__ xn,
                            const float* __restrict__ wi, const float* __restrict__ wf,
                            const float* __restrict__ wz, const float* __restrict__ wo,
                            const float* __restrict__ bi, const float* __restrict__ bf,
                            const float* __restrict__ bz, const float* __restrict__ bo,
                            float* __restrict__ pre, int T) {
  int idx = blockIdx.x * blockDim.x + threadIdx.x;
  if (idx >= T * 40) return;
  int t = idx / 40, e = idx % 40;
  int blk = e / DHS, o = e % DHS;
  const float* xr = xc + (size_t)t * 40 + blk * DHS;
  const float* nr = xn + (size_t)t * 40 + blk * DHS;
  float ai = bi[e], af = bf[e], az = bz[e], ao = bo[e];
#pragma unroll
  for (int i2 = 0; i2 < DHS; ++i2) {
    int wix = blk * 25 + i2 * DHS + o;
    float xv = xr[i2], nv = nr[i2];
    ai += xv * wi[wix]; af += xv * wf[wix];
    az += nv * wz[wix]; ao += nv * wo[wix];
  }
  float* pr = pre + (size_t)t * 160;
  pr[e] = ai; pr[40 + e] = af; pr[80 + e] = az; pr[120 + e] = ao;
}

// one thread per (batch,head): register-resident state, LDS-cached R mats
__global__ void k_slstm_scan(const float* __restrict__ pre,
                             const float* __restrict__ ri, const float* __restrict__ rf,
                             const float* __restrict__ rz, const float* __restrict__ ro,
                             float* __restrict__ hs, int nBH) {
  __shared__ float Rs[4 * NHS * 25];
  for (int i = threadIdx.x; i < 4 * NHS * 25; i += blockDim.x) {
    int g = i / (NHS * 25), rmd = i % (NHS * 25);
    const float* Rp = (g == 0) ? ri : (g == 1) ? rf : (g == 2) ? rz : ro;
    Rs[i] = Rp[rmd];
  }
  __syncthreads();
  int idx = blockIdx.x * blockDim.x + threadIdx.x;
  if (idx >= nBH) return;
  int b = idx / NHS, hd = idx % NHS;
  float h[DHS], c[DHS], nn[DHS], m[DHS];
  for (int e = 0; e < DHS; ++e) { h[e] = 0.f; c[e] = 0.f; nn[e] = 0.f; m[e] = 0.f; }
  for (int s = 0; s < SEQ; ++s) {
    const float* pb = pre + (size_t)(b * SEQ + s) * 160 + hd * DHS;
    float ir[DHS], fr[DHS], zr[DHS], orr[DHS];
    for (int e = 0; e < DHS; ++e) {
      float a0 = 0.f, a1 = 0.f, a2 = 0.f, a3 = 0.f;
#pragma unroll
      for (int d = 0; d < DHS; ++d) {
        int off = hd * 25 + d * DHS + e;
        a0 += h[d] * Rs[0 * NHS * 25 + off];
        a1 += h[d] * Rs[1 * NHS * 25 + off];
        a2 += h[d] * Rs[2 * NHS * 25 + off];
        a3 += h[d] * Rs[3 * NHS * 25 + off];
      }
      ir[e] = pb[e] + a0; fr[e] = pb[40 + e] + a1;
      zr[e] = pb[80 + e] + a2; orr[e] = pb[120 + e] + a3;
    }
    for (int e = 0; e < DHS; ++e) {
      float mn = fmaxf(fr[e] + m[e], ir[e]);
      float ig = expf(ir[e] - mn);
      float fg = expf(fr[e] + m[e] - mn);
      float cn = fg * c[e] + ig * tanhf(zr[e]);
      float nnew = fg * nn[e] + ig;
      float hn = (1.f / (1.f + expf(-orr[e]))) * cn / fmaxf(nnew, 1e-6f);
      c[e] = cn; nn[e] = nnew; m[e] = mn; h[e] = hn;
    }
    float* hrow = hs + (size_t)(b * SEQ + s) * 40 + hd * DHS;
    for (int e = 0; e < DHS; ++e) hrow[e] = h[e];
  }
}

// h[t,:] += per-head(DH=5) LayerNorm(hs)*w
__global__ void k_mhnorm_addres(const float* __restrict__ hs, const float* __restrict__ w,
                                float* __restrict__ h, int T) {
  int idx = blockIdx.x * blockDim.x + threadIdx.x;
  if (idx >= T * NHS) return;
  int t = idx / NHS, hd = idx % NHS;
  const float* r = hs + (size_t)t * 40 + hd * DHS;
  float loc[DHS];
  float mu = 0.f;
  for (int d = 0; d < DHS; ++d) { loc[d] = r[d]; mu += loc[d]; }
  mu *= (1.0f / DHS);
  float var = 0.f;
  for (int d = 0; d < DHS; ++d) { float dd = loc[d] - mu; var += dd * dd; }
  var *= (1.0f / DHS);
  float inv = rsqrtf(var + 1e-5f);
  for (int d = 0; d < DHS; ++d) {
    int cc = hd * DHS + d;
    h[(size_t)t * 40 + cc] += (loc[d] - mu) * inv * w[cc];
  }
}

// GeGLU (tanh-approx gelu, matching jax.nn.gelu default)
__global__ void k_geglu(const float* __restrict__ u, float* __restrict__ gv, int T) {
  int idx = blockIdx.x * blockDim.x + threadIdx.x;
  if (idx >= T * 64) return;
  int t = idx / 64, j = idx % 64;
  float g = u[(size_t)t * 128 + j];
  float v = u[(size_t)t * 128 + 64 + j];
  float gg = 0.5f * g * (1.0f + tanhf(0.7978845608028654f * (g + 0.044715f * g * g * g)));
  gv[(size_t)t * 64 + j] = gg * v;
}

__global__ void k_out(const float* __restrict__ h, const float* __restrict__ ow,
                      const float* __restrict__ ob, float* __restrict__ out, int T) {
  int t = blockIdx.x * blockDim.x + threadIdx.x;
  if (t >= T) return;
  const float* r = h + (size_t)t * 40;
  float acc = ob[0];
  for (int e = 0; e < 40; ++e) acc += r[e] * ow[e];
  out[t] = acc;
}

// ------------------------------- host side ---------------------------------
struct MB {
  const float *ln_w, *up_w, *conv_w, *conv_b, *q_w, *k_w, *v_w,
              *ig_w, *ig_b, *fg_w, *fg_b, *mh_norm_w, *skip, *down_w;
};
struct SB {
  const float *ln_w, *conv_w, *conv_b, *wi, *wf, *wz, *wo, *ri, *rf, *rz, *ro,
              *bi, *bf, *bz, *bo, *mh_norm_w, *ln2_w,
              *ff_up_w, *ff_up_b, *ff_down_w, *ff_down_b;
};

static inline int cdiv(int a, int b) { return (a + b - 1) / b; }

extern "C" void kernel_launch(void* const* d_in, const int* in_sizes, int n_in,
                              void* d_out, int out_size, void* d_ws, size_t ws_size,
                              hipStream_t stream) {
  const float* P[128];
  for (int i = 0; i < n_in && i < 128; ++i) P[i] = (const float*)d_in[i];

  const float *x = nullptr, *in_w = nullptr, *in_b = nullptr,
              *post_w = nullptr, *out_w = nullptr, *out_b = nullptr;
  MB mb[3]; SB sb;

  bool insertion = (n_in > 0 && in_sizes[0] == 4096 * 24 * 10);
  int i = 0;
  if (insertion) {
    x = P[i++]; in_w = P[i++]; in_b = P[i++]; post_w = P[i++]; out_w = P[i++]; out_b = P[i++];
    auto rdM = [&](MB& m) {
      m.ln_w = P[i++]; m.up_w = P[i++]; m.conv_w = P[i++]; m.conv_b = P[i++];
      m.q_w = P[i++]; m.k_w = P[i++]; m.v_w = P[i++];
      m.ig_w = P[i++]; m.ig_b = P[i++]; m.fg_w = P[i++]; m.fg_b = P[i++];
      m.mh_norm_w = P[i++]; m.skip = P[i++]; m.down_w = P[i++];
    };
    auto rdS = [&](SB& s) {
      s.ln_w = P[i++]; s.conv_w = P[i++]; s.conv_b = P[i++];
      s.wi = P[i++]; s.wf = P[i++]; s.wz = P[i++]; s.wo = P[i++];
      s.ri = P[i++]; s.rf = P[i++]; s.rz = P[i++]; s.ro = P[i++];
      s.bi = P[i++]; s.bf = P[i++]; s.bz = P[i++]; s.bo = P[i++];
      s.mh_norm_w = P[i++]; s.ln2_w = P[i++];
      s.ff_up_w = P[i++]; s.ff_up_b = P[i++]; s.ff_down_w = P[i++]; s.ff_down_b = P[i++];
    };
    rdM(mb[0]); rdS(sb); rdM(mb[1]); rdM(mb[2]);
  } else {
    // jax pytree order: sorted dict keys, 'blocks' list first inside params, x last
    auto rdM = [&](MB& m) {
      m.conv_b = P[i++]; m.conv_w = P[i++]; m.down_w = P[i++];
      m.fg_b = P[i++]; m.fg_w = P[i++]; m.ig_b = P[i++]; m.ig_w = P[i++];
      m.k_w = P[i++]; m.ln_w = P[i++]; m.mh_norm_w = P[i++];
      m.q_w = P[i++]; m.skip = P[i++]; m.up_w = P[i++]; m.v_w = P[i++];
    };
    auto rdS = [&](SB& s) {
      s.bf = P[i++]; s.bi = P[i++]; s.bo = P[i++]; s.bz = P[i++];
      s.conv_b = P[i++]; s.conv_w = P[i++];
      s.ff_down_b = P[i++]; s.ff_down_w = P[i++]; s.ff_up_b = P[i++]; s.ff_up_w = P[i++];
      s.ln2_w = P[i++]; s.ln_w = P[i++]; s.mh_norm_w = P[i++];
      s.rf = P[i++]; s.ri = P[i++]; s.ro = P[i++]; s.rz = P[i++];
      s.wf = P[i++]; s.wi = P[i++]; s.wo = P[i++]; s.wz = P[i++];
    };
    rdM(mb[0]); rdS(sb); rdM(mb[1]); rdM(mb[2]);
    in_b = P[i++]; in_w = P[i++]; out_b = P[i++]; out_w = P[i++]; post_w = P[i++]; x = P[i++];
  }

  // batch chunking so scratch fits (all coupling is within a batch)
  int CB = 512;
  while (CB > 2 && (size_t)CB * SEQ * 656 * sizeof(float) > ws_size) CB >>= 1;
  int nch = 4096 / CB;
  int T = CB * SEQ;

  float* W0 = (float*)d_ws;
  float* Hc = W0;
  float* T0 = Hc + (size_t)T * 40;
  float* T1 = T0 + (size_t)T * 40;    // up / pre / ffn-up (ld 160 region)
  float* T2 = T1 + (size_t)T * 160;   // xc / gv
  float* T3 = T2 + (size_t)T * 80;    // qkv
  float* T4 = T3 + (size_t)T * 240;   // gates
  float* T5 = T4 + (size_t)T * 16;    // hatt/h80 / hs

  auto gemm = [&](const float* A, const float* Wt, const float* bias, const float* res,
                  float* D, int M, int K, int N) {
    int Nt = cdiv(N, 16);
    int waves = (M / 16) * Nt;
    k_gemm<<<cdiv(waves, 8), 256, 0, stream>>>(A, Wt, bias, res, D, M, K, N, Nt);
  };

  for (int ch = 0; ch < nch; ++ch) {
    const float* xch = x + (size_t)ch * T * 10;
    float* outch = (float*)d_out + (size_t)ch * T;

    k_input<<<cdiv(T * 40, 256), 256, 0, stream>>>(xch, in_w, in_b, Hc, T);

    for (int blk = 0; blk < 4; ++blk) {
      if (blk == 1) {
        // ---- sLSTM block + FFN ----
        k_layernorm<<<cdiv(T, 256), 256, 0, stream>>>(Hc, sb.ln_w, T0, T, 40);
        k_conv_silu<<<cdiv(T * 40, 256), 256, 0, stream>>>(T0, 40, sb.conv_w, sb.conv_b, T2, T, 40);
        k_slstm_pre<<<cdiv(T * 40, 256), 256, 0, stream>>>(T2, T0, sb.wi, sb.wf, sb.wz, sb.wo,
                                                           sb.bi, sb.bf, sb.bz, sb.bo, T1, T);
        k_slstm_scan<<<cdiv(CB * NHS, 256), 256, 0, stream>>>(T1, sb.ri, sb.rf, sb.rz, sb.ro,
                                                              T5, CB * NHS);
        k_mhnorm_addres<<<cdiv(T * NHS, 256), 256, 0, stream>>>(T5, sb.mh_norm_w, Hc, T);
        k_layernorm<<<cdiv(T, 256), 256, 0, stream>>>(Hc, sb.ln2_w, T0, T, 40);
        gemm(T0, sb.ff_up_w, sb.ff_up_b, nullptr, T1, T, 40, 128);
        k_geglu<<<cdiv(T * 64, 256), 256, 0, stream>>>(T1, T2, T);
        gemm(T2, sb.ff_down_w, sb.ff_down_b, Hc, Hc, T, 64, 40);
      } else {
        // ---- mLSTM block ----
        const MB& m = mb[(blk == 0) ? 0 : blk - 1];
        k_layernorm<<<cdiv(T, 256), 256, 0, stream>>>(Hc, m.ln_w, T0, T, 40);
        gemm(T0, m.up_w, nullptr, nullptr, T1, T, 40, 160);
        k_conv_silu<<<cdiv(T * 80, 256), 256, 0, stream>>>(T1, 160, m.conv_w, m.conv_b, T2, T, 80);
        k_headwise_qkv<<<cdiv(T * 80, 256), 256, 0, stream>>>(T2, T1, m.q_w, m.k_w, m.v_w, T3, T);
        {
          int waves = T / 16;
          k_gemm_gates<<<cdiv(waves, 8), 256, 0, stream>>>(T3, m.ig_w, m.fg_w, m.ig_b, m.fg_b, T4, T);
        }
        k_attn<<<CB * NHM, 32, 0, stream>>>(T3, T4, T5);
        k_mhnorm_gate<<<cdiv(T * NHM, 256), 256, 0, stream>>>(T5, m.mh_norm_w, m.skip, T2, T1, T5, T);
        gemm(T5, m.down_w, nullptr, Hc, Hc, T, 80, 40);
      }
    }

    k_layernorm<<<cdiv(T, 256), 256, 0, stream>>>(Hc, post_w, T0, T, 40);
    k_out<<<cdiv(T, 256), 256, 0, stream>>>(T0, out_w, out_b, outch, T);
  }
}